// GCNWMMSE_SISOAdhoc_81750407512804
// MI455X (gfx1250) — compile-verified
//
#include <hip/hip_runtime.h>

// MI455X / gfx1250 implementation of GCN-WMMSE SISO-adhoc layers.
// B=64 batches, K=512 users, C=4, 5 shared-parameter layers.
// Strategy: precompute Habs2 (67MB, fits in 192MB L2) into workspace, then one
// persistent workgroup per batch runs all 5 layers; both GEMV passes per layer
// are done with V_WMMA_F32_16X16X4_F32 (full f32 precision on matrix pipe).

typedef float v2f __attribute__((ext_vector_type(2)));
typedef float v8f __attribute__((ext_vector_type(8)));

#define KDIM 512
#define BDIM 64
#define NLAYERS 5

// ---------------------------------------------------------------------------
// Kernel 1: Habs2[i] = h_re[i]^2 + h_im[i]^2   (streams 134MB in, 67MB out)
// ---------------------------------------------------------------------------
__global__ __launch_bounds__(256) void habs2_kernel(
    const float* __restrict__ hre, const float* __restrict__ him,
    float* __restrict__ habs2, long nvec4) {
  long i = (long)blockIdx.x * blockDim.x + threadIdx.x;
  const long stride = (long)gridDim.x * blockDim.x;
  for (; i < nvec4; i += stride) {
    const float4 r = ((const float4*)hre)[i];
    const float4 m = ((const float4*)him)[i];
    float4 o;
    o.x = r.x * r.x + m.x * m.x;
    o.y = r.y * r.y + m.y * m.y;
    o.z = r.z * r.z + m.z * m.z;
    o.w = r.w * r.w + m.w * m.w;
    ((float4*)habs2)[i] = o;
  }
}

// ---------------------------------------------------------------------------
// Kernel 2: persistent per-batch WMMSE layers.
// 1024 threads = 32 waves; each wave owns one 16-row (or 16-col) WMMA tile.
// FUSED=true computes Habs2 on the fly from h_re/h_im (workspace too small).
// ---------------------------------------------------------------------------
template <bool FUSED>
__global__ __launch_bounds__(1024, 1) void wmmse_kernel(
    const float* __restrict__ Aglob,   // habs2 [B,K,K] if !FUSED
    const float* __restrict__ hre, const float* __restrict__ him,
    const float* __restrict__ noise, const float* __restrict__ maxp,
    const float* __restrict__ vre0, const float* __restrict__ vim0,
    const float* __restrict__ arre, const float* __restrict__ arim,
    const float* __restrict__ dlre, const float* __restrict__ dlim,
    const float* __restrict__ rcre, const float* __restrict__ rcim,
    float* __restrict__ out) {
  __shared__ __align__(16) float sP[KDIM];    // |v|^2 (pass-1 vector)
  __shared__ __align__(16) float sVec[KDIM];  // GEMV result (cov raw / t)
  __shared__ __align__(16) float sQ[KDIM];    // |u|^2 |w| (pass-2 vector)
  __shared__ __align__(16) float sVr[KDIM];
  __shared__ __align__(16) float sVi[KDIM];
  __shared__ __align__(16) float sVtr[KDIM];
  __shared__ __align__(16) float sVti[KDIM];

  const int b = blockIdx.x;
  const int tid = threadIdx.x;
  const int lane = tid & 31;   // wave32 on gfx1250
  const int wave = tid >> 5;   // 0..31
  const long base = (long)b * KDIM * KDIM;

  const float* __restrict__ Ab = FUSED ? nullptr : (Aglob + base);
  const float* __restrict__ Rb = hre + base;
  const float* __restrict__ Ib = him + base;

  // broadcast tiny complex params into registers (C=4)
  float ar_r[4], ar_i[4], dl_r[4], dl_i[4], rc_r[4], rc_i[4];
#pragma unroll
  for (int c = 0; c < 4; ++c) {
    ar_r[c] = arre[c]; ar_i[c] = arim[c];
    dl_r[c] = dlre[c]; dl_i[c] = dlim[c];
    rc_r[c] = rcre[c]; rc_i[c] = rcim[c];
  }

  float dr = 0.f, di = 0.f, nz = 0.f, P = 1.f, sqrtP = 1.f;
  if (tid < KDIM) {
    dr = hre[base + (long)tid * (KDIM + 1)];  // diag(H)
    di = him[base + (long)tid * (KDIM + 1)];
    nz = noise[b * KDIM + tid];
    P = maxp[b * KDIM + tid];
    sqrtP = sqrtf(P);
    sVr[tid] = vre0[b * KDIM + tid];
    sVi[tid] = vim0[b * KDIM + tid];
  }
  const float dabs2 = dr * dr + di * di;
  __syncthreads();

  const int koff = (lane < 16) ? 0 : 2;  // K split across wave halves
  const int mrow = lane & 15;

  for (int L = 0; L < NLAYERS; ++L) {
    // ---- phase 1: p = |v|^2 ----
    if (tid < KDIM) {
      float a = sVr[tid], c = sVi[tid];
      sP[tid] = a * a + c * c;
    }
    __syncthreads();

    // ---- phase 2: cov_raw = Habs2 * p  (row GEMV via WMMA) ----
    {
      const int r0 = wave * 16;
      v8f acc = {};
      const long rowbase = (long)(r0 + mrow) * KDIM;
#pragma unroll 4
      for (int c0 = 0; c0 < KDIM; c0 += 4) {
        v2f afrag;  // A-tile: A[r0+m, c0+k] in ISA A-layout
        if (!FUSED) {
          afrag = *(const v2f*)(Ab + rowbase + c0 + koff);
        } else {
          v2f rr = *(const v2f*)(Rb + rowbase + c0 + koff);
          v2f ii = *(const v2f*)(Ib + rowbase + c0 + koff);
          afrag.x = rr.x * rr.x + ii.x * ii.x;
          afrag.y = rr.y * rr.y + ii.y * ii.y;
        }
        // B-tile: p[c0+k] broadcast across all 16 columns
        v2f bfrag = *(const v2f*)(sP + c0 + koff);
        acc = __builtin_amdgcn_wmma_f32_16x16x4_f32(
            false, afrag, false, bfrag, (short)0, acc, false, false);
      }
      // every D column identical; lane0 holds rows r0..r0+7, lane16 rows +8
      if (lane == 0) {
#pragma unroll
        for (int m = 0; m < 8; ++m) sVec[r0 + m] = acc[m];
      } else if (lane == 16) {
#pragma unroll
        for (int m = 0; m < 8; ++m) sVec[r0 + 8 + m] = acc[m];
      }
    }
    __syncthreads();

    // ---- phase 3: u/w/v_tilde/q (err is analytically real) ----
    if (tid < KDIM) {
      float c = sVec[tid] + nz;        // cov
      float vr = sVr[tid], vi = sVi[tid];
      float gr = dr * vr - di * vi;    // g = diagH * v
      float gi = dr * vi + di * vr;
      float g2 = gr * gr + gi * gi;    // |g|^2  (u = g/c)
      float invc = 1.0f / c;
      float er = fmaf(-g2, invc, 1.0f);  // err = 1 - |g|^2/c  (real)
      float w = 1.0f / er;
      float s = dabs2 * w * invc;      // v_tilde = |d|^2 w/c * v
      sVtr[tid] = s * vr;
      sVti[tid] = s * vi;
      sQ[tid] = g2 * invc * invc * fabsf(w);  // |u|^2 |w|
    }
    __syncthreads();

    // ---- phase 4: t = Habs2^T * q  (transpose GEMV via WMMA) ----
    {
      const int n0 = wave * 16;
      v8f acc = {};
#pragma unroll 4
      for (int j0 = 0; j0 < KDIM; j0 += 4) {
        v2f bfrag;  // B-tile: natural 4x16 tile A[j0+k, n0+n] (coalesced)
        const long off0 = (long)(j0 + koff) * KDIM + n0 + mrow;
        if (!FUSED) {
          bfrag.x = Ab[off0];
          bfrag.y = Ab[off0 + KDIM];
        } else {
          float r0v = Rb[off0], i0v = Ib[off0];
          float r1v = Rb[off0 + KDIM], i1v = Ib[off0 + KDIM];
          bfrag.x = r0v * r0v + i0v * i0v;
          bfrag.y = r1v * r1v + i1v * i1v;
        }
        // A-tile: q[j0+k] broadcast across all 16 rows
        v2f afrag = *(const v2f*)(sQ + j0 + koff);
        acc = __builtin_amdgcn_wmma_f32_16x16x4_f32(
            false, afrag, false, bfrag, (short)0, acc, false, false);
      }
      // every D row identical; lane l (<16) col n0+l lives in acc[0]
      if (lane < 16) sVec[n0 + lane] = acc[0];
    }
    __syncthreads();

    // ---- phase 5: mu/ula, ar+dl expand, CReLU, rc recombine, power corr ----
    if (tid < KDIM) {
      float ul = fabsf(sVec[tid]);
      float vtr = sVtr[tid], vti = sVti[tid];
      float vta = sqrtf(vtr * vtr + vti * vti);
      float mu = fmaxf(vta / sqrtP - ul, 0.0f);
      float ula = ul + mu;
      float inv_ula = 1.0f / ula;
      float z1r = vtr * inv_ula, z1i = vti * inv_ula;
      float nr = 0.f, ni = 0.f;
#pragma unroll
      for (int c = 0; c < 4; ++c) {
        float tr = z1r * ar_r[c] - z1i * ar_i[c] + vtr * dl_r[c] - vti * dl_i[c];
        float ti = z1r * ar_i[c] + z1i * ar_r[c] + vtr * dl_i[c] + vti * dl_r[c];
        tr = fmaxf(tr, 0.0f);  // complex ReLU
        ti = fmaxf(ti, 0.0f);
        nr += tr * rc_r[c] - ti * rc_i[c];
        ni += tr * rc_i[c] + ti * rc_r[c];
      }
      float pw = nr * nr + ni * ni;
      float cur = fmaxf(P, pw);
      float corr = fminf(sqrtf(P / cur), 1.0f);
      nr *= corr;
      ni *= corr;
      sVr[tid] = nr;
      sVi[tid] = ni;
      long oidx = (((long)L * BDIM + b) * KDIM + tid) * 2;  // [L,B,K,1,2]
      out[oidx] = nr;
      out[oidx + 1] = ni;
    }
    __syncthreads();
  }
}

// ---------------------------------------------------------------------------
extern "C" void kernel_launch(void* const* d_in, const int* in_sizes, int n_in,
                              void* d_out, int out_size, void* d_ws,
                              size_t ws_size, hipStream_t stream) {
  const float* hre = (const float*)d_in[0];
  const float* him = (const float*)d_in[1];
  const float* noise = (const float*)d_in[2];
  const float* maxp = (const float*)d_in[3];
  const float* vre = (const float*)d_in[4];
  const float* vim = (const float*)d_in[5];
  const float* arre = (const float*)d_in[6];
  const float* arim = (const float*)d_in[7];
  const float* dlre = (const float*)d_in[8];
  const float* dlim = (const float*)d_in[9];
  const float* rcre = (const float*)d_in[10];
  const float* rcim = (const float*)d_in[11];
  float* out = (float*)d_out;

  const long nA = (long)BDIM * KDIM * KDIM;
  const size_t habsBytes = (size_t)nA * sizeof(float);

  if (ws_size >= habsBytes) {
    // Materialize |H|^2 once (67MB, L2-resident for the 10 GEMV passes).
    float* habs2 = (float*)d_ws;
    habs2_kernel<<<4096, 256, 0, stream>>>(hre, him, habs2, nA / 4);
    wmmse_kernel<false><<<BDIM, 1024, 0, stream>>>(
        habs2, hre, him, noise, maxp, vre, vim, arre, arim, dlre, dlim, rcre,
        rcim, out);
  } else {
    // Workspace too small: fuse |H|^2 computation into the GEMVs.
    wmmse_kernel<true><<<BDIM, 1024, 0, stream>>>(
        nullptr, hre, him, noise, maxp, vre, vim, arre, arim, dlre, dlim, rcre,
        rcim, out);
  }
}